// TTFloquetBlock_82497731821664
// MI455X (gfx1250) — compile-verified
//
#include <hip/hip_runtime.h>

// y[b,s,d] = x[b,s,d] * gain[d]
// gain[d]  = sum_{n<M} cos(n*OMEGA*DT - eps[n,d]*DT),  eps = fe * (1 + 0.1*dw[n])
// Memory-bound streaming op: 537 MB HBM traffic, ~23 us floor at 23.3 TB/s.

typedef float v4f __attribute__((ext_vector_type(4)));

static constexpr float DT    = 0.01f;
static constexpr float OMEGA = 1.0f;

// ---------------------------------------------------------------------------
// Kernel 1: gain[d] (tiny: M*D = 65K cos evals). One thread per d.
// ---------------------------------------------------------------------------
__global__ __launch_bounds__(256) void floquet_gain_kernel(
    const float* __restrict__ fe,   // [M, D]
    const float* __restrict__ dw,   // [M]
    float* __restrict__ gain,       // [D]
    int M, int D) {
  const int d = blockIdx.x * blockDim.x + threadIdx.x;
  if (d >= D) return;
  float acc = 0.0f;
  for (int n = 0; n < M; ++n) {
    const float scale = 1.0f + 0.1f * dw[n];
    const float eps   = fe[(size_t)n * (size_t)D + d] * scale;
    acc += __cosf((float)n * (OMEGA * DT) - eps * DT);
  }
  gain[d] = acc;
}

// ---------------------------------------------------------------------------
// Kernel 2: y = x * gain[d], streamed.
// Each block owns a 1024-float column tile. Its 4 KB gain slice is staged
// into LDS once with GLOBAL_LOAD_ASYNC_TO_LDS_B128 (ASYNCcnt-tracked CDNA5
// async copy), then read back via ds_load_b128 into registers. The row loop
// is 8-deep unrolled: NT B128 loads -> v_pk_mul -> NT B128 stores, with a
// global_prefetch_b8 group covering exactly the next unroll iteration.
// ---------------------------------------------------------------------------
__global__ __launch_bounds__(256) void floquet_scale_kernel(
    const v4f* __restrict__ x,      // [R, d4] as float4
    const float* __restrict__ gain, // [D]
    v4f* __restrict__ y,            // [R, d4]
    long long d4,                   // D / 4
    long long R) {                  // B * S rows
  __shared__ __align__(16) float s_gain[256 * 4];   // 4 KB slice

  const long long col4  = (long long)blockIdx.x * 256 + threadIdx.x;
  const long long col4c = (col4 < d4) ? col4 : (d4 - 1);   // clamp for staging

  // ---- Stage gain slice: global -> LDS via async DMA (gfx1250 path). ----
  {
    // Low 32 bits of a generic LDS pointer are the LDS byte offset
    // (generic LDS addr = {shared_aperture_hi, lds_offset}, ISA 10.2).
    unsigned lds_off = (unsigned)(unsigned long long)(&s_gain[threadIdx.x * 4]);
    const float* gsrc = gain + col4c * 4;   // 16B-aligned (d_ws is 256B-aligned)
    asm volatile("global_load_async_to_lds_b128 %0, %1, off"
                 :
                 : "v"(lds_off), "v"(gsrc)
                 : "memory");
#if defined(__has_builtin) && __has_builtin(__builtin_amdgcn_s_wait_asynccnt)
    __builtin_amdgcn_s_wait_asynccnt(0);
#else
    asm volatile("s_wait_asynccnt 0x0" ::: "memory");
#endif
  }
  __syncthreads();

  const v4f g = *(const v4f*)&s_gain[threadIdx.x * 4];  // ds_load_b128

  if (col4 >= d4) return;
  const long long rows_per = (R + gridDim.y - 1) / (long long)gridDim.y;
  const long long r0 = (long long)blockIdx.y * rows_per;
  if (r0 >= R) return;
  const long long nrows = min(r0 + rows_per, R) - r0;

  const v4f* __restrict__ xr = x + r0 * d4 + col4;
  v4f* __restrict__ yr       = y + r0 * d4 + col4;

  long long r = 0;
  for (; r + 8 <= nrows; r += 8) {
    // Prefetch exactly the next unroll group (rows r+8..r+15). Speculative:
    // past-the-end prefetches are dropped on translation failure.
#pragma unroll
    for (int k = 0; k < 8; ++k)
      __builtin_prefetch(xr + (8 + k) * d4, 0, 0);

    v4f v[8];
#pragma unroll
    for (int k = 0; k < 8; ++k)
      v[k] = __builtin_nontemporal_load(xr + k * d4);
#pragma unroll
    for (int k = 0; k < 8; ++k)
      v[k] *= g;
#pragma unroll
    for (int k = 0; k < 8; ++k)
      __builtin_nontemporal_store(v[k], yr + k * d4);

    xr += 8 * d4;
    yr += 8 * d4;
  }
  for (; r < nrows; ++r) {
    v4f a = __builtin_nontemporal_load(xr);
    __builtin_nontemporal_store(a * g, yr);
    xr += d4;
    yr += d4;
  }
}

// ---------------------------------------------------------------------------
extern "C" void kernel_launch(void* const* d_in, const int* in_sizes, int n_in,
                              void* d_out, int out_size, void* d_ws, size_t ws_size,
                              hipStream_t stream) {
  const float* x  = (const float*)d_in[0];  // [B,S,D]
  const float* fe = (const float*)d_in[1];  // [M,D]
  const float* dw = (const float*)d_in[2];  // [M]
  // d_in[3] = coupling_matrix: unused by the reference forward.
  float* y    = (float*)d_out;
  float* gain = (float*)d_ws;               // D floats = 16 KB scratch

  const int M = in_sizes[2];
  const int D = in_sizes[1] / M;            // 4096
  const long long N = (long long)in_sizes[0];
  const long long R = N / D;                // B*S rows (16384)

  // Phase 1: gain table (stream-ordered before phase 2).
  floquet_gain_kernel<<<dim3((D + 255) / 256), dim3(256), 0, stream>>>(
      fe, dw, gain, M, D);

  // Phase 2: streaming scale. D assumed multiple of 4 (reference: 4096).
  const long long d4 = D / 4;
  const unsigned gx = (unsigned)((d4 + 255) / 256);          // 4 column tiles
  unsigned gy = (unsigned)((R < 512) ? (R > 0 ? R : 1) : 512);
  floquet_scale_kernel<<<dim3(gx, gy), dim3(256), 0, stream>>>(
      (const v4f*)x, gain, (v4f*)y, d4, R);
}